// VectorQuantizerEMA_6768868459198
// MI455X (gfx1250) — compile-verified
//
#include <hip/hip_runtime.h>

// ---------------- problem constants ----------------
#define NUM_EMB   1024
#define DIM       64
#define NROWS     65536            // 64 * 1024 rows
#define DECAY     0.99f
#define EPS       1e-5f
#define CCOST     0.25f

// ---------------- output layout (floats, concatenated in return order) ----
#define OUT_LOSS    0
#define OUT_Q       1                          // quantized_ste == quantized (STE identity)
#define OUT_IDX     (OUT_Q + NROWS * DIM)      // 4194305
#define OUT_NEW_W   (OUT_IDX + NROWS)          // 4259841
#define OUT_NCS     (OUT_NEW_W + NUM_EMB*DIM)  // 4325377
#define OUT_NEW_EMA (OUT_NCS + NUM_EMB)        // 4326401

// ---------------- workspace layout (floats) ----------------
#define WS_COUNTS 0
#define WS_DW     (WS_COUNTS + NUM_EMB)
#define WS_LOSS   (WS_DW + NUM_EMB * DIM)
#define WS_ENORM  (WS_LOSS + 32)               // padded, 16B aligned
#define WS_ZERO_N (WS_LOSS + 1)                // region that must be zeroed per launch

// dynamic LDS: codebook (256KB) + ||e||^2 (4KB) = 266240 bytes (<320KB/WGP)
#define LDS_EN_OFF  (NUM_EMB * DIM)
#define LDS_BYTES   ((NUM_EMB * DIM + NUM_EMB) * 4)

typedef __attribute__((ext_vector_type(2))) float v2f;
typedef __attribute__((ext_vector_type(8))) float v8f;

// ---------------------------------------------------------------
// Kernel 0: zero the accumulation scratch (counts, dw, loss)
// ---------------------------------------------------------------
__global__ void vq_zero_ws(float* __restrict__ ws, int n) {
    int i = blockIdx.x * blockDim.x + threadIdx.x;
    if (i < n) ws[i] = 0.0f;
}

// ---------------------------------------------------------------
// Kernel 1: ||e_k||^2 per codebook entry
// ---------------------------------------------------------------
__global__ void vq_enorm(const float* __restrict__ E, float* __restrict__ ws) {
    int k = blockIdx.x * blockDim.x + threadIdx.x;
    if (k < NUM_EMB) {
        float s = 0.0f;
        #pragma unroll
        for (int d = 0; d < DIM; ++d) { float v = E[k * DIM + d]; s += v * v; }
        ws[WS_ENORM + k] = s;
    }
}

// ---------------------------------------------------------------
// Kernel 2: main WMMA distance + argmin + scatter kernel.
// Codebook staged in LDS via gfx1250 async-to-LDS (ASYNCcnt) path;
// B fragments double-buffered in registers over ds_load_b64 so the
// chained v_wmma_f32_16x16x4_f32 pipe never waits on memory.
// ---------------------------------------------------------------
__global__ __launch_bounds__(256) void vq_main(const float* __restrict__ X,
                                               const float* __restrict__ E,
                                               float* __restrict__ out,
                                               float* __restrict__ ws) {
    extern __shared__ float lds_E[];           // [NUM_EMB*DIM] codebook + [NUM_EMB] enorm
    float* lds_EN = lds_E + LDS_EN_OFF;

    const int lane = threadIdx.x & 31;
    const int wave = threadIdx.x >> 5;
    const int half = lane >> 4;      // 0: lanes 0-15, 1: lanes 16-31
    const int l15  = lane & 15;
    const int row0 = (blockIdx.x * 8 + wave) * 16;

    // ---- stage codebook (256KB) + enorm (4KB) into LDS asynchronously ----
    {
        // low 32 bits of a generic LDS pointer = wave-relative LDS byte offset
        unsigned lbase = (unsigned)(size_t)(void*)lds_E;
        const char* gsrc = (const char*)E;
        #pragma unroll 8
        for (int i = 0; i < (NUM_EMB * DIM * 4) / (256 * 16); ++i) {
            unsigned off = (unsigned)threadIdx.x * 16u + (unsigned)i * 4096u;
            unsigned ldst = lbase + off;
            const char* gp = gsrc + off;
            asm volatile("global_load_async_to_lds_b128 %0, %1, off"
                         :: "v"(ldst), "v"(gp) : "memory");
        }
        // 4KB of enorm: one b128 per thread
        {
            unsigned off = (unsigned)threadIdx.x * 16u;
            unsigned ldst = lbase + (unsigned)(LDS_EN_OFF * 4) + off;
            const char* gp = (const char*)(ws + WS_ENORM) + off;
            asm volatile("global_load_async_to_lds_b128 %0, %1, off"
                         :: "v"(ldst), "v"(gp) : "memory");
        }
        asm volatile("s_wait_asynccnt 0x0" ::: "memory");
        __syncthreads();
    }

    // ---- A fragments: ISA 32-bit A 16x4 layout ----
    // lane<16 holds (M=l15, K=k0+{0,1}); lane>=16 holds (M=l15, K=k0+{2,3})
    v2f a[16];
    const float* xrow = X + (row0 + l15) * DIM + 2 * half;
    #pragma unroll
    for (int j = 0; j < 16; ++j)
        a[j] = *(const v2f*)(xrow + 4 * j);

    float minv[8];
    int   mini[8];
    #pragma unroll
    for (int r = 0; r < 8; ++r) { minv[r] = 3.0e38f; mini[r] = 0; }

    // process one 16-code tile given its resident B fragments
    auto process = [&](int t, const v2f* b) {
        v8f acc = {};
        #pragma unroll
        for (int j = 0; j < 16; ++j)
            acc = __builtin_amdgcn_wmma_f32_16x16x4_f32(
                false, a[j], false, b[j], (short)0, acc, false, false);
        // acc[r] @ lane = dot(row0 + r + 8*half, 16*t + l15)
        const float en   = lds_EN[t * 16 + l15];
        const int   code = t * 16 + l15;
        #pragma unroll
        for (int r = 0; r < 8; ++r) {
            float val = en - 2.0f * acc[r];
            if (val < minv[r]) { minv[r] = val; mini[r] = code; }
        }
    };

    // ---- double-buffered B fragments from LDS (mirror of A layout) ----
    v2f b0[16], b1[16];
    {
        const float* er = lds_E + l15 * DIM + 2 * half;
        #pragma unroll
        for (int j = 0; j < 16; ++j) b0[j] = *(const v2f*)(er + 4 * j);
    }
    for (int ct = 0; ct < NUM_EMB / 16; ct += 2) {
        {   // prefetch tile ct+1 while tile ct runs through the WMMA pipe
            const float* er = lds_E + ((ct + 1) * 16 + l15) * DIM + 2 * half;
            #pragma unroll
            for (int j = 0; j < 16; ++j) b1[j] = *(const v2f*)(er + 4 * j);
        }
        process(ct, b0);
        if (ct + 2 < NUM_EMB / 16) {   // prefetch tile ct+2
            const float* er = lds_E + ((ct + 2) * 16 + l15) * DIM + 2 * half;
            #pragma unroll
            for (int j = 0; j < 16; ++j) b0[j] = *(const v2f*)(er + 4 * j);
        }
        process(ct + 1, b1);
    }

    // ---- cross-lane argmin within each 16-lane half (rows differ per half) ----
    #pragma unroll
    for (int r = 0; r < 8; ++r) {
        #pragma unroll
        for (int off = 8; off >= 1; off >>= 1) {
            float ov = __shfl_xor(minv[r], off, 32);
            int   oi = __shfl_xor(mini[r], off, 32);
            if (ov < minv[r] || (ov == minv[r] && oi < mini[r])) {
                minv[r] = ov; mini[r] = oi;
            }
        }
    }

    // lanes 0 (rows row0+r) and 16 (rows row0+8+r) hold the winners
    if (l15 == 0) {
        #pragma unroll
        for (int r = 0; r < 8; ++r) {
            int row = row0 + r + 8 * half;
            out[OUT_IDX + row] = (float)mini[r];
            atomicAdd(&ws[WS_COUNTS + mini[r]], 1.0f);
        }
    }

    // ---- quantized gather (from LDS) + commitment loss + dw segment-sum ----
    float lsum = 0.0f;
    #pragma unroll
    for (int r = 0; r < 16; ++r) {
        int idx = __shfl(mini[r & 7], (r < 8) ? 0 : 16, 32);
        int row = row0 + r;
        v2f q = *(const v2f*)(lds_E + idx * DIM + 2 * lane);
        v2f x = *(const v2f*)(X + (size_t)row * DIM + 2 * lane);
        // out base is offset by 1 float (loss slot) -> scalar stores (4B aligned)
        float* qo = out + OUT_Q + (size_t)row * DIM + 2 * lane;
        qo[0] = q.x; qo[1] = q.y;
        float dx = x.x - q.x, dy = x.y - q.y;
        lsum += dx * dx + dy * dy;
        atomicAdd(&ws[WS_DW + idx * DIM + 2 * lane],     x.x);
        atomicAdd(&ws[WS_DW + idx * DIM + 2 * lane + 1], x.y);
    }
    #pragma unroll
    for (int off = 16; off >= 1; off >>= 1) lsum += __shfl_xor(lsum, off, 32);
    if (lane == 0) atomicAdd(&ws[WS_LOSS], lsum);
}

// ---------------------------------------------------------------
// Kernel 3: EMA update + normalization (one 1024-thread workgroup)
// ---------------------------------------------------------------
__global__ __launch_bounds__(1024) void vq_finalize(const float* __restrict__ ema_cs,
                                                    const float* __restrict__ ema_w,
                                                    float* __restrict__ out,
                                                    const float* __restrict__ ws) {
    __shared__ float red[1024];
    const int k = threadIdx.x;
    float ncs = ema_cs[k] * DECAY + (1.0f - DECAY) * ws[WS_COUNTS + k];
    out[OUT_NCS + k] = ncs;
    red[k] = ncs;
    __syncthreads();
    for (int s = 512; s > 0; s >>= 1) {
        if (k < s) red[k] += red[k + s];
        __syncthreads();
    }
    const float n = red[0];
    const float norm = (ncs + EPS) / (n + NUM_EMB * EPS) * n;
    #pragma unroll 4
    for (int d = 0; d < DIM; ++d) {
        float nw = ema_w[k * DIM + d] * DECAY + (1.0f - DECAY) * ws[WS_DW + k * DIM + d];
        out[OUT_NEW_EMA + k * DIM + d] = nw;
        out[OUT_NEW_W   + k * DIM + d] = nw / norm;
    }
    if (k == 0) out[OUT_LOSS] = CCOST * ws[WS_LOSS] / (float)((size_t)NROWS * DIM);
}

// ---------------------------------------------------------------
extern "C" void kernel_launch(void* const* d_in, const int* in_sizes, int n_in,
                              void* d_out, int out_size, void* d_ws, size_t ws_size,
                              hipStream_t stream) {
    const float* X   = (const float*)d_in[0];  // inputs [64,1024,64]
    const float* E   = (const float*)d_in[1];  // emb_weight [1024,64]
    const float* ecs = (const float*)d_in[2];  // ema_cluster_size [1024]
    const float* ew  = (const float*)d_in[3];  // ema_w [1024,64]
    float* out = (float*)d_out;
    float* ws  = (float*)d_ws;

    (void)in_sizes; (void)n_in; (void)out_size; (void)ws_size;

    vq_zero_ws<<<(WS_ZERO_N + 255) / 256, 256, 0, stream>>>(ws, WS_ZERO_N);
    vq_enorm<<<NUM_EMB / 256, 256, 0, stream>>>(E, ws);
    vq_main<<<NROWS / (16 * 8), 256, LDS_BYTES, stream>>>(X, E, out, ws);
    vq_finalize<<<1, 1024, 0, stream>>>(ecs, ew, out, ws);
}